// MeteoGraphPC_v3_67035849556159
// MI455X (gfx1250) — compile-verified
//
#include <hip/hip_runtime.h>

#define DEV __device__ __forceinline__

constexpr int Bc = 2, Tc = 8, Nn = 4096, Ec = 32768;
constexpr int FIN = 16, DEc = 8, Hc = 128, HDS = 4, NTc = 4, LGc = 3, LTc = 4;
constexpr int HH = HDS * Hc; // 512

typedef __attribute__((ext_vector_type(8))) float v8f;
typedef __attribute__((ext_vector_type(4))) float v4f;
typedef __attribute__((ext_vector_type(2))) float v2f;

DEV v8f wmma4(v2f a, v2f b, v8f c) {
  return __builtin_amdgcn_wmma_f32_16x16x4_f32(false, a, false, b, (short)0, c,
                                               false, false);
}

// ---------------------------------------------------------------------------
// LDS-tiled WMMA f32 16x16x4 GEMM.
// Block = 4 waves = 32 rows x 128 cols. Each wave owns a 32x32 macro-tile
// (2M x 2N C tiles, 4 accumulators) -> 4 v_wmma per k-step.
// Per K-chunk (KC = min(Kd,32)) the block cooperatively stages
//   A[32][KC]  (coalesced b128 loads)   -> 4 KB LDS
//   B[KC][128] (coalesced b128 loads)   -> 16 KB LDS
// and all fragments are then read from LDS (ds_load_b64 / ds_load_b32).
// Fragment layouts per CDNA5 ISA 7.12.2 (wave32):
//   A 16x4 f32 : per lane the contiguous pair A[m, ka], A[m, ka+1], ka=k0+2*kh
//   B 4x16 f32 : per lane W[ka, n], W[ka+1, n]
//   C/D 16x16  : VGPR r -> rows r (lanes0-15) / r+8 (lanes16-31), N = lane&15
// ---------------------------------------------------------------------------
template <int Kd>
__global__ void gemm_wmma(const float* __restrict__ A, int lda,
                          const float* __restrict__ W, int ldw,
                          const float* __restrict__ bias,
                          float* __restrict__ C, int ldc) {
  constexpr int KC = (Kd < 32) ? Kd : 32;
  __shared__ __align__(16) float sA[32 * KC];
  __shared__ __align__(16) float sB[KC * 128];
  const int lane = threadIdx.x, m = lane & 15, kh = lane >> 4;
  const int wv = threadIdx.y;           // 0..3
  const int tid = wv * 32 + lane;       // 0..127
  const int row0 = blockIdx.x * 32;
  const int colB = blockIdx.y * 128;    // block column slab
  const int colW = wv * 32;             // wave strip inside slab
  __builtin_prefetch(A + (size_t)row0 * lda, 0, 1);
  v8f c00 = {0.f,0.f,0.f,0.f,0.f,0.f,0.f,0.f};
  v8f c01 = c00, c10 = c00, c11 = c00;
  for (int kb = 0; kb < Kd; kb += KC) {
    // ---- stage A[32][KC] ----
    for (int i = tid; i < (32 * KC) / 4; i += 128) {
      int fl = i * 4, r = fl / KC, kk = fl % KC;
      *(v4f*)&sA[fl] = *(const v4f*)(A + (size_t)(row0 + r) * lda + kb + kk);
    }
    // ---- stage B[KC][128] ----
    for (int i = tid; i < (KC * 128) / 4; i += 128) {
      int fl = i * 4, kk = fl / 128, cc = fl % 128;
      *(v4f*)&sB[fl] = *(const v4f*)(W + (size_t)(kb + kk) * ldw + colB + cc);
    }
    __syncthreads();
#pragma unroll
    for (int k0 = 0; k0 < KC; k0 += 4) {
      const int ka = k0 + 2 * kh;
      v2f a0 = *(const v2f*)&sA[m * KC + ka];
      v2f a1 = *(const v2f*)&sA[(m + 16) * KC + ka];
      v2f b0, b1;
      b0.x = sB[ka * 128 + colW + m];
      b0.y = sB[(ka + 1) * 128 + colW + m];
      b1.x = sB[ka * 128 + colW + 16 + m];
      b1.y = sB[(ka + 1) * 128 + colW + 16 + m];
      c00 = wmma4(a0, b0, c00);
      c01 = wmma4(a0, b1, c01);
      c10 = wmma4(a1, b0, c10);
      c11 = wmma4(a1, b1, c11);
    }
    __syncthreads();
  }
  const int cg0 = colB + colW + m, cg1 = cg0 + 16;
  float bb0 = bias ? bias[cg0] : 0.f;
  float bb1 = bias ? bias[cg1] : 0.f;
#pragma unroll
  for (int r = 0; r < 8; ++r) {
    size_t ra = (size_t)(row0 + r + 8 * kh) * ldc;
    size_t rb = (size_t)(row0 + 16 + r + 8 * kh) * ldc;
    C[ra + cg0] = c00[r] + bb0;
    C[ra + cg1] = c01[r] + bb1;
    C[rb + cg0] = c10[r] + bb0;
    C[rb + cg1] = c11[r] + bb1;
  }
}

// ---------------------------------------------------------------------------
// Dilated temporal conv: 3 shifted K=128 GEMMs, same LDS tiling as above.
// X,Y: [B,T,N,H], row=(b,t,n). Wc: [O][I][K]. 'SAME' padding: output t reads
// t+(k-1)*dil; invalid taps skipped (validity is uniform per block since a
// 32-row tile never crosses a (b,t) boundary). The weight stage transposes
// [O][I][tap] -> sB[kk][o] with coalesced LDS stores.
// ---------------------------------------------------------------------------
__global__ void tconv_wmma(const float* __restrict__ X, const float* __restrict__ Wc,
                           const float* __restrict__ bias, float* __restrict__ Y,
                           int dil) {
  __shared__ __align__(16) float sA[32 * 32];
  __shared__ __align__(16) float sB[32 * 128];
  const int lane = threadIdx.x, m = lane & 15, kh = lane >> 4;
  const int wv = threadIdx.y;
  const int tid = wv * 32 + lane;
  const int row0 = blockIdx.x * 32;
  const int colW = wv * 32;
  const int bt = row0 / Nn, n0 = row0 - bt * Nn, t = bt % Tc, btb = bt - t;
  v8f c00 = {0.f,0.f,0.f,0.f,0.f,0.f,0.f,0.f};
  v8f c01 = c00, c10 = c00, c11 = c00;
  for (int tap = 0; tap < 3; ++tap) {
    int t2 = t + (tap - 1) * dil;
    if (t2 < 0 || t2 >= Tc) continue; // zero padding, block-uniform
    const float* Abase = X + ((size_t)(btb + t2) * Nn + n0) * Hc;
    __builtin_prefetch(Abase, 0, 1);
    for (int kb = 0; kb < Hc; kb += 32) {
      for (int i = tid; i < (32 * 32) / 4; i += 128) {
        int fl = i * 4, r = fl / 32, kk = fl % 32;
        *(v4f*)&sA[fl] = *(const v4f*)(Abase + (size_t)r * Hc + kb + kk);
      }
      for (int i = tid; i < 32 * 128; i += 128) {
        int kk = i / 128, o = i % 128; // coalesced LDS store, strided gather
        sB[kk * 128 + o] = Wc[((size_t)o * Hc + kb + kk) * 3 + tap];
      }
      __syncthreads();
#pragma unroll
      for (int k0 = 0; k0 < 32; k0 += 4) {
        const int ka = k0 + 2 * kh;
        v2f a0 = *(const v2f*)&sA[m * 32 + ka];
        v2f a1 = *(const v2f*)&sA[(m + 16) * 32 + ka];
        v2f b0, b1;
        b0.x = sB[ka * 128 + colW + m];
        b0.y = sB[(ka + 1) * 128 + colW + m];
        b1.x = sB[ka * 128 + colW + 16 + m];
        b1.y = sB[(ka + 1) * 128 + colW + 16 + m];
        c00 = wmma4(a0, b0, c00);
        c01 = wmma4(a0, b1, c01);
        c10 = wmma4(a1, b0, c10);
        c11 = wmma4(a1, b1, c11);
      }
      __syncthreads();
    }
  }
  const int cg0 = colW + m, cg1 = cg0 + 16;
  float bb0 = bias[cg0];
  float bb1 = bias[cg1];
#pragma unroll
  for (int r = 0; r < 8; ++r) {
    size_t ra = (size_t)(row0 + r + 8 * kh) * Hc;
    size_t rb = (size_t)(row0 + 16 + r + 8 * kh) * Hc;
    Y[ra + cg0] = c00[r] + bb0;
    Y[ra + cg1] = c01[r] + bb1;
    Y[rb + cg0] = c10[r] + bb0;
    Y[rb + cg1] = c11[r] + bb1;
  }
}

DEV float blk_sum(float v, float* sb, int c) {
  __syncthreads();
  sb[c] = v;
  __syncthreads();
  for (int s = Hc >> 1; s > 0; s >>= 1) {
    if (c < s) sb[c] += sb[c + s];
    __syncthreads();
  }
  float r = sb[0];
  __syncthreads();
  return r;
}

// LayerNorm over 128 channels per row + exact GELU (TCN stage).
__global__ void tcn_lngelu(const float* __restrict__ in, const float* __restrict__ g,
                           const float* __restrict__ bt, float* __restrict__ out) {
  __shared__ float sb[Hc];
  int row = blockIdx.x, c = threadIdx.x;
  float x = in[(size_t)row * Hc + c];
  float mu = blk_sum(x, sb, c) * (1.f / Hc);
  float d = x - mu;
  float var = blk_sum(d * d, sb, c) * (1.f / Hc);
  float yv = d * rsqrtf(var + 1e-5f) * g[c] + bt[c];
  out[(size_t)row * Hc + c] = 0.5f * yv * (1.f + erff(yv * 0.70710678118654752f));
}

// out = LN(tmp + res) with gamma/beta, per row of 128.
__global__ void resid_ln(const float* __restrict__ tmp, const float* __restrict__ res,
                         const float* __restrict__ g, const float* __restrict__ bt,
                         float* __restrict__ out) {
  __shared__ float sb[Hc];
  int row = blockIdx.x, c = threadIdx.x;
  float x = tmp[(size_t)row * Hc + c] + res[(size_t)row * Hc + c];
  float mu = blk_sum(x, sb, c) * (1.f / Hc);
  float d = x - mu;
  float var = blk_sum(d * d, sb, c) * (1.f / Hc);
  out[(size_t)row * Hc + c] = d * rsqrtf(var + 1e-5f) * g[c] + bt[c];
}

__global__ void fill_f32(float* p, float v, long long n) {
  long long i = (long long)blockIdx.x * blockDim.x + threadIdx.x;
  if (i < n) p[i] = v;
}
__global__ void fill_u32(unsigned* p, unsigned v, long long n) {
  long long i = (long long)blockIdx.x * blockDim.x + threadIdx.x;
  if (i < n) p[i] = v;
}

// deg[n] = 1 (self loop) + in-degree over dst
__global__ void deg_count(const int* __restrict__ ei, float* __restrict__ deg) {
  int e = blockIdx.x * blockDim.x + threadIdx.x;
  if (e >= Ec) return;
  atomicAdd(&deg[ei[Ec + e]], 1.f);
}

// g[b,dst,c] += xw[b,src,c] * dinv[src]*dinv[dst]
__global__ void gcn_scatter(const int* __restrict__ ei, const float* __restrict__ deg,
                            const float* __restrict__ xw, float* __restrict__ g) {
  int tid = blockIdx.x * blockDim.x + threadIdx.x;
  if (tid >= Bc * Ec * Hc) return;
  int c = tid % Hc, e = (tid / Hc) % Ec, b = tid / (Hc * Ec);
  int src = ei[e], dst = ei[Ec + e];
  float nrm = rsqrtf(deg[src]) * rsqrtf(deg[dst]);
  atomicAdd(&g[((size_t)b * Nn + dst) * Hc + c],
            xw[((size_t)b * Nn + src) * Hc + c] * nrm);
}

// g = relu(g + xw/deg (self loop) + bias)
__global__ void gcn_finish(const float* __restrict__ deg, const float* __restrict__ xw,
                           const float* __restrict__ bias, float* __restrict__ g) {
  int tid = blockIdx.x * blockDim.x + threadIdx.x;
  if (tid >= Bc * Nn * Hc) return;
  int c = tid % Hc, n = (tid / Hc) % Nn;
  float v = g[tid] + xw[tid] * (1.f / deg[n]) + bias[c];
  g[tid] = fmaxf(v, 0.f);
}

DEV unsigned fkey(float f) {
  unsigned u = __float_as_uint(f);
  return (u & 0x80000000u) ? ~u : (u | 0x80000000u);
}
DEV float fdec(unsigned u) {
  return (u & 0x80000000u) ? __uint_as_float(u & 0x7fffffffu)
                           : __uint_as_float(~u);
}

// alpha[b,e,h] = <q[dst,h,:], q[src,h,:]+e[e,h,:]>/sqrt(H); segment-max via atomic key.
__global__ void alpha_amax(const int* __restrict__ ei, const float* __restrict__ q,
                           const float* __restrict__ eb, float* __restrict__ alpha,
                           unsigned* __restrict__ amk) {
  int tid = blockIdx.x * blockDim.x + threadIdx.x;
  if (tid >= Bc * Ec * HDS) return;
  int h = tid % HDS, e = (tid / HDS) % Ec, b = tid / (HDS * Ec);
  int src = ei[e], dst = ei[Ec + e];
  const float* qs = q + ((size_t)b * Nn + src) * HH + h * Hc;
  const float* qd = q + ((size_t)b * Nn + dst) * HH + h * Hc;
  const float* ee = eb + (size_t)e * HH + h * Hc;
  float s = 0.f;
  for (int i = 0; i < Hc; ++i) s += qd[i] * (qs[i] + ee[i]);
  s *= 0.088388347648318447f; // 1/sqrt(128)
  alpha[tid] = s;
  atomicMax(&amk[((size_t)b * Nn + dst) * HDS + h], fkey(s));
}

// decode keys; empty segments (-inf) -> 0 as in reference where(isfinite,...,0)
__global__ void amax_fix(const unsigned* __restrict__ amk, float* __restrict__ amx) {
  int tid = blockIdx.x * blockDim.x + threadIdx.x;
  if (tid >= Bc * Nn * HDS) return;
  unsigned u = amk[tid];
  float v = (u == 0u) ? 0.f : fdec(u);
  if (!(v > -3.0e38f && v < 3.0e38f)) v = 0.f;
  amx[tid] = v;
}

// alpha -> exp(alpha - amax[dst]); den[dst] += ex
__global__ void exp_den(const int* __restrict__ ei, const float* __restrict__ amx,
                        float* __restrict__ alpha, float* __restrict__ den) {
  int tid = blockIdx.x * blockDim.x + threadIdx.x;
  if (tid >= Bc * Ec * HDS) return;
  int h = tid % HDS, e = (tid / HDS) % Ec, b = tid / (HDS * Ec);
  int dst = ei[Ec + e];
  float ex = expf(alpha[tid] - amx[((size_t)b * Nn + dst) * HDS + h]);
  alpha[tid] = ex;
  atomicAdd(&den[((size_t)b * Nn + dst) * HDS + h], ex);
}

// acc[b,dst,hc] += (q[src,hc] + e[e,hc]) * ex/(den[dst]+1e-16)
__global__ void acc_scatter(const int* __restrict__ ei, const float* __restrict__ q,
                            const float* __restrict__ eb, const float* __restrict__ ex,
                            const float* __restrict__ den, float* __restrict__ acc) {
  int tid = blockIdx.x * blockDim.x + threadIdx.x;
  if (tid >= Bc * Ec * HH) return;
  int hc = tid % HH, e = (tid / HH) % Ec, b = tid / (HH * Ec);
  int h = hc >> 7;
  int src = ei[e], dst = ei[Ec + e];
  float w = ex[((size_t)b * Ec + e) * HDS + h] /
            (den[((size_t)b * Nn + dst) * HDS + h] + 1e-16f);
  atomicAdd(&acc[((size_t)b * Nn + dst) * HH + hc],
            (q[((size_t)b * Nn + src) * HH + hc] + eb[(size_t)e * HH + hc]) * w);
}

// Final LN over H, head 128->4, mask.
__global__ void final_head(const float* __restrict__ hin, const float* __restrict__ g,
                           const float* __restrict__ bt, const float* __restrict__ hw,
                           const float* __restrict__ hb, const float* __restrict__ mask,
                           float* __restrict__ y) {
  __shared__ float sb[Hc];
  __shared__ float sx[Hc];
  int row = blockIdx.x, c = threadIdx.x;
  float x = hin[(size_t)row * Hc + c];
  float mu = blk_sum(x, sb, c) * (1.f / Hc);
  float d = x - mu;
  float var = blk_sum(d * d, sb, c) * (1.f / Hc);
  sx[c] = d * rsqrtf(var + 1e-5f) * g[c] + bt[c];
  __syncthreads();
  if (c < NTc) {
    float s = hb[c];
    for (int i = 0; i < Hc; ++i) s += sx[i] * hw[i * NTc + c];
    y[(size_t)row * NTc + c] = s * mask[row];
  }
}

extern "C" void kernel_launch(void* const* d_in, const int* in_sizes, int n_in,
                              void* d_out, int out_size, void* d_ws, size_t ws_size,
                              hipStream_t stream) {
  (void)in_sizes; (void)n_in; (void)out_size; (void)ws_size;
  const float* x_seq  = (const float*)d_in[0];
  const int*   ei_seq = (const int*)d_in[1];
  const float* ea_seq = (const float*)d_in[2];
  const float* mask   = (const float*)d_in[3];
  const float* in_W   = (const float*)d_in[4];
  const float* in_b   = (const float*)d_in[5];
  const float* conv_W = (const float*)d_in[6];
  const float* conv_b = (const float*)d_in[7];
  const float* tcn_g  = (const float*)d_in[8];
  const float* tcn_be = (const float*)d_in[9];
  const float* gcn_W  = (const float*)d_in[10];
  const float* gcn_b  = (const float*)d_in[11];
  const float* a_Wn   = (const float*)d_in[12];
  const float* a_We   = (const float*)d_in[13];
  const float* a_Wo   = (const float*)d_in[14];
  const float* a_bo   = (const float*)d_in[15];
  const float* a_g    = (const float*)d_in[16];
  const float* a_be   = (const float*)d_in[17];
  const float* lng    = (const float*)d_in[18];
  const float* lnb    = (const float*)d_in[19];
  const float* hd_W   = (const float*)d_in[20];
  const float* hd_b   = (const float*)d_in[21];
  float* y = (float*)d_out;

  // workspace carve-up (~190 MB, all re-initialized every call)
  float* ws = (float*)d_ws;
  size_t off = 0;
  auto alloc = [&](size_t n) { float* p = ws + off; off += n; return p; };
  float* h0   = alloc((size_t)Bc * Tc * Nn * Hc);   // [B,T,N,H]
  float* h1   = alloc((size_t)Bc * Tc * Nn * Hc);   // [B,T,N,H]
  float* ebuf = alloc((size_t)Ec * HH);             // edge proj per (t,layer)
  float* qbuf = alloc((size_t)Bc * Nn * HH);
  float* accb = alloc((size_t)Bc * Nn * HH);
  float* xw   = alloc((size_t)Bc * Nn * Hc);
  float* gA   = alloc((size_t)Bc * Nn * Hc);        // gcn output (residual src)
  float* gB   = alloc((size_t)Bc * Nn * Hc);        // layer output ping buffer
  float* tmpb = alloc((size_t)Bc * Nn * Hc);
  float* deg  = alloc((size_t)Nn);
  float* alf  = alloc((size_t)Bc * Ec * HDS);
  unsigned* amk = (unsigned*)(void*)alloc((size_t)Bc * Nn * HDS);
  float* amx  = alloc((size_t)Bc * Nn * HDS);
  float* den  = alloc((size_t)Bc * Nn * HDS);

  auto nb = [](long long n) { return (unsigned)((n + 255) / 256); };
  auto fillf = [&](float* p, float v, long long n) {
    fill_f32<<<nb(n), 256, 0, stream>>>(p, v, n);
  };
  auto fillu = [&](unsigned* p, unsigned v, long long n) {
    fill_u32<<<nb(n), 256, 0, stream>>>(p, v, n);
  };

  // 1) input projection: h0 = x_seq @ in_W + in_b     (M=65536, N=128, K=16)
  gemm_wmma<FIN><<<dim3(Bc * Tc * Nn / 32, Hc / 128), dim3(32, 4), 0, stream>>>(
      x_seq, FIN, in_W, Hc, in_b, h0, Hc);

  // 2) temporal dilated conv stack (conv h0->h1, LN+GELU h1->h0)
  for (int l = 0; l < LTc; ++l) {
    tconv_wmma<<<dim3(Bc * Tc * Nn / 32, 1), dim3(32, 4), 0, stream>>>(
        h0, conv_W + (size_t)l * Hc * Hc * 3, conv_b + l * Hc, h1, 1 << l);
    tcn_lngelu<<<Bc * Tc * Nn, Hc, 0, stream>>>(h1, tcn_g + l * Hc,
                                                tcn_be + l * Hc, h0);
  }

  // 3) per-timestep GCN + attention stack (timesteps independent)
  for (int t = 0; t < Tc; ++t) {
    const int* ei = ei_seq + (size_t)t * 2 * Ec;
    fillf(deg, 1.0f, Nn); // self loops
    deg_count<<<Ec / 256, 256, 0, stream>>>(ei, deg);
    const float* inP[2] = {h0 + (size_t)(0 * Tc + t) * Nn * Hc,
                           h0 + (size_t)(1 * Tc + t) * Nn * Hc};
    for (int l = 0; l < LGc; ++l) {
      // e = edge_attr[t] @ We[l]                      (M=32768, N=512, K=8)
      gemm_wmma<DEc><<<dim3(Ec / 32, HH / 128), dim3(32, 4), 0, stream>>>(
          ea_seq + (size_t)t * Ec * DEc, DEc, a_We + (size_t)l * DEc * HH, HH,
          nullptr, ebuf, HH);
      // xw = x @ gcn_W[l]                             (M=4096, N=128, K=128)
      for (int b = 0; b < Bc; ++b)
        gemm_wmma<Hc><<<dim3(Nn / 32, Hc / 128), dim3(32, 4), 0, stream>>>(
            inP[b], Hc, gcn_W + (size_t)l * Hc * Hc, Hc, nullptr,
            xw + (size_t)b * Nn * Hc, Hc);
      fillf(gA, 0.f, (long long)Bc * Nn * Hc);
      gcn_scatter<<<(Bc * Ec * Hc) / 256, 256, 0, stream>>>(ei, deg, xw, gA);
      gcn_finish<<<(Bc * Nn * Hc) / 256, 256, 0, stream>>>(deg, xw,
                                                           gcn_b + l * Hc, gA);
      // q = g @ Wn[l]                                 (M=4096, N=512, K=128)
      for (int b = 0; b < Bc; ++b)
        gemm_wmma<Hc><<<dim3(Nn / 32, HH / 128), dim3(32, 4), 0, stream>>>(
            gA + (size_t)b * Nn * Hc, Hc, a_Wn + (size_t)l * Hc * HH, HH,
            nullptr, qbuf + (size_t)b * Nn * HH, HH);
      fillu(amk, 0u, (long long)Bc * Nn * HDS);
      fillf(den, 0.f, (long long)Bc * Nn * HDS);
      fillf(accb, 0.f, (long long)Bc * Nn * HH);
      alpha_amax<<<(Bc * Ec * HDS) / 256, 256, 0, stream>>>(ei, qbuf, ebuf, alf, amk);
      amax_fix<<<(Bc * Nn * HDS) / 256, 256, 0, stream>>>(amk, amx);
      exp_den<<<(Bc * Ec * HDS) / 256, 256, 0, stream>>>(ei, amx, alf, den);
      acc_scatter<<<(Bc * Ec * HH) / 256, 256, 0, stream>>>(ei, qbuf, ebuf, alf,
                                                            den, accb);
      const float* nextP[2];
      for (int b = 0; b < Bc; ++b) {
        // o = acc @ Wo[l] + bo                        (M=4096, N=128, K=512)
        gemm_wmma<HH><<<dim3(Nn / 32, Hc / 128), dim3(32, 4), 0, stream>>>(
            accb + (size_t)b * Nn * HH, HH, a_Wo + (size_t)l * HH * Hc, Hc,
            a_bo + l * Hc, tmpb + (size_t)b * Nn * Hc, Hc);
        float* outP = (l == LGc - 1) ? (h1 + (size_t)(b * Tc + t) * Nn * Hc)
                                     : (gB + (size_t)b * Nn * Hc);
        resid_ln<<<Nn, Hc, 0, stream>>>(tmpb + (size_t)b * Nn * Hc,
                                        gA + (size_t)b * Nn * Hc, a_g + l * Hc,
                                        a_be + l * Hc, outP);
        nextP[b] = outP;
      }
      inP[0] = nextP[0];
      inP[1] = nextP[1];
    }
  }

  // 4) final LN + head + mask
  final_head<<<Bc * Tc * Nn, Hc, 0, stream>>>(h1, lng, lnb, hd_W, hd_b, mask, y);
}